// dynamical_system_49520972922979
// MI455X (gfx1250) — compile-verified
//
#include <hip/hip_runtime.h>
#include <stdint.h>

// Lorenz RK4, B=16384 particles, NT=2000 steps, out shape (B,3,2001) fp32.
// Compute: per-thread fp32 RK4 chain (3-way ILP from x/y/z).
// Writeback: staged in LDS, drained by gfx1250 async LDS->global DMA
// (global_store_async_from_lds_b32, ASYNCcnt), double-buffered so the
// 393 MB trajectory write fully overlaps the dependent compute chain.

#define B_THREADS 64          // 2 wave32s per workgroup -> many WGPs busy
#define NT_STEPS  2000
#define CHUNK     16          // steps buffered per LDS tile
#define NCHUNKS   (NT_STEPS / CHUNK)   // 125
#define TRAJ      2001
#define DT_F      0.01f

// 48 async b32 stores per wave per chunk; wait <=48 keeps exactly one
// chunk of DMA in flight (double buffer) while the next chunk computes.
#define ASYNC_PER_CHUNK 48

// Emit one async LDS->global store per time-slot, with the slot offset as an
// immediate. ISA (CDNA5 ISA): INST_OFFSET is added to BOTH the
// global address (GVS mode) and the LDS address,
// so one immediate walks both sides of the row.
template <int TL>
struct AsyncRowStore {
  static __device__ __forceinline__ void go(uint32_t vaddr, uint32_t laddr,
                                            uint64_t sbase) {
    AsyncRowStore<TL - 1>::go(vaddr, laddr, sbase);
    asm volatile("global_store_async_from_lds_b32 %0, %1, %2 offset:%3"
                 :
                 : "v"(vaddr), "v"(laddr), "s"(sbase), "n"((TL - 1) * 4)
                 : "memory");
  }
};
template <>
struct AsyncRowStore<0> {
  static __device__ __forceinline__ void go(uint32_t, uint32_t, uint64_t) {}
};

__global__ __launch_bounds__(B_THREADS)
void lorenz_rk4_async(const float* __restrict__ x0,
                      const float* __restrict__ prm,
                      float* __restrict__ out, int nPart) {
  // [buf][wave][coord][lane][17] : 17-float pitch => LDS rows stride 17
  // dwords; gcd(17,64)=1 so both the per-step ds_store pattern (lane-strided)
  // and the async engine's per-instruction reads are bank-conflict free.
  __shared__ float lds[2][B_THREADS / 32][3][32][17];  // 26112 B

  const int tid = blockIdx.x * B_THREADS + threadIdx.x;  // particle id
  if (tid >= nPart) return;                              // full blocks: no divergence
  const int w = threadIdx.x >> 5;
  const int l = threadIdx.x & 31;

  const float p0 = prm[0], p1 = prm[1], p2 = prm[2];
  float x = x0[3 * tid + 0];
  float y = x0[3 * tid + 1];
  float z = x0[3 * tid + 2];

  // t = 0 slot is the initial condition.
  const uint64_t row0 = (uint64_t)tid * 3u * (uint64_t)TRAJ;
  out[row0 + 0u * TRAJ] = x;
  out[row0 + 1u * TRAJ] = y;
  out[row0 + 2u * TRAJ] = z;

  // Per-lane 32-bit byte offsets of the three (particle,coord) rows
  // (max 393 MB < 2^31, fits signed 32-bit GVS VADDR).
  const uint32_t vb0 = (uint32_t)(row0 * 4u);
  const uint32_t vb1 = vb0 + TRAJ * 4u;
  const uint32_t vb2 = vb1 + TRAJ * 4u;

  // LDS byte addresses of this lane's three rows in buffer 0 (low 32 bits of
  // the generic pointer are the LDS offset on AMDGPU).
  const uint32_t lr0 = (uint32_t)(uintptr_t)&lds[0][w][0][l][0];
  const uint32_t lr1 = (uint32_t)(uintptr_t)&lds[0][w][1][l][0];
  const uint32_t lr2 = (uint32_t)(uintptr_t)&lds[0][w][2][l][0];
  const uint32_t bufStride = (uint32_t)sizeof(lds[0]);  // 13056 B

  // Uniform SGPR base for GVS addressing: out + 4 * t0 (t0 starts at 1).
  uint64_t sbase = (uint64_t)(uintptr_t)out + 4u;

  const float dt = DT_F, h2 = 0.5f * DT_F, h6 = DT_F / 6.0f;

  int buf = 0;
  for (int ch = 0; ch < NCHUNKS; ++ch) {
    // Buffer 'buf' was drained once at most ASYNC_PER_CHUNK stores (the
    // previous chunk's) remain in flight -> previous DMA overlaps this
    // chunk's compute; the chunk before that is guaranteed complete.
    asm volatile("s_wait_asynccnt 48" ::: "memory");

#pragma unroll
    for (int tl = 0; tl < CHUNK; ++tl) {
      // Classic RK4 on the Lorenz RHS (fp32, compiler contracts to FMA).
      float k1x = p0 * (y - x);
      float k1y = x * (p1 - z) - y;
      float k1z = x * y - p2 * z;
      float ax = x + h2 * k1x, ay = y + h2 * k1y, az = z + h2 * k1z;
      float k2x = p0 * (ay - ax);
      float k2y = ax * (p1 - az) - ay;
      float k2z = ax * ay - p2 * az;
      float bx = x + h2 * k2x, by = y + h2 * k2y, bz = z + h2 * k2z;
      float k3x = p0 * (by - bx);
      float k3y = bx * (p1 - bz) - by;
      float k3z = bx * by - p2 * bz;
      float cx = x + dt * k3x, cy = y + dt * k3y, cz = z + dt * k3z;
      float k4x = p0 * (cy - cx);
      float k4y = cx * (p1 - cz) - cy;
      float k4z = cx * cy - p2 * cz;
      x += h6 * (k1x + 2.0f * (k2x + k3x) + k4x);
      y += h6 * (k1y + 2.0f * (k2y + k3y) + k4y);
      z += h6 * (k1z + 2.0f * (k2z + k3z) + k4z);

      lds[buf][w][0][l][tl] = x;
      lds[buf][w][1][l][tl] = y;
      lds[buf][w][2][l][tl] = z;
    }

    // Make this wave's LDS writes visible to the async engine. Each wave
    // only drains LDS it wrote itself, so no workgroup barrier is needed.
    asm volatile("s_wait_dscnt 0" ::: "memory");

    const uint32_t bofs = buf ? bufStride : 0u;
    AsyncRowStore<CHUNK>::go(vb0, lr0 + bofs, sbase);
    AsyncRowStore<CHUNK>::go(vb1, lr1 + bofs, sbase);
    AsyncRowStore<CHUNK>::go(vb2, lr2 + bofs, sbase);

    sbase += (uint64_t)(CHUNK * 4);
    buf ^= 1;
  }

  // Drain remaining DMA before wave end (s_endpgm also implies wait-idle).
  asm volatile("s_wait_asynccnt 0" ::: "memory");
}

extern "C" void kernel_launch(void* const* d_in, const int* in_sizes, int n_in,
                              void* d_out, int out_size, void* d_ws,
                              size_t ws_size, hipStream_t stream) {
  (void)n_in; (void)out_size; (void)d_ws; (void)ws_size;
  const float* x0  = (const float*)d_in[0];  // (B,3) fp32
  const float* prm = (const float*)d_in[1];  // (3,)  fp32
  float* out = (float*)d_out;                // (B,3,2001) fp32

  const int nPart = in_sizes[0] / 3;         // 16384
  const int blocks = (nPart + B_THREADS - 1) / B_THREADS;
  hipLaunchKernelGGL(lorenz_rk4_async, dim3(blocks), dim3(B_THREADS), 0,
                     stream, x0, prm, out, nPart);
}